// FCOSPostProcessor_58377195487236
// MI455X (gfx1250) — compile-verified
//
#include <hip/hip_runtime.h>
#include <math.h>

// ---------------- problem constants (match reference setup) ----------------
#define NN     16
#define CC     80
#define HH     100
#define WW     152
#define HWSZ   (HH * WW)           // 15200
#define TOTAL  ((long long)NN * CC * HWSZ)  // 19,456,000
#define KTOP   1000
#define KP     1008                 // padded to 63 tiles of 16
#define NTILE  (KP / 16)            // 63
#define CAP    8192                 // per-image candidate capacity
#define OUT_DETS (NN * 100 * 5)     // 8000 floats, labels follow
#define PRE_LOGIT (-2.9444389791664403f)  // logit(0.05)
#define NMS_T  0.6f

typedef __attribute__((ext_vector_type(2))) float v2f;
typedef __attribute__((ext_vector_type(8))) float v8f;

__device__ __forceinline__ float sigf(float x) { return 1.0f / (1.0f + __expf(-x)); }

// ---------------- 0: zero scratch (hist + cnt + mask region) ----------------
__global__ void k_zero(unsigned* __restrict__ p, long long count) {
  long long i = (long long)blockIdx.x * blockDim.x + threadIdx.x;
  if (i < count) p[i] = 0u;
}

// ---------------- 1: streaming scan -> per-image 2048-bucket histogram ------
__global__ void k_scan_hist(const float* __restrict__ cls,
                            const float* __restrict__ ctr,
                            unsigned* __restrict__ hist) {
  __shared__ unsigned h[2048];
  for (int i = threadIdx.x; i < 2048; i += blockDim.x) h[i] = 0u;
  __syncthreads();
  long long e = (long long)blockIdx.x * blockDim.x + threadIdx.x;  // < TOTAL
  const long long per = (long long)CC * HWSZ;   // block never straddles images
  int n  = (int)(e / per);
  long long rem = e - (long long)n * per;
  int hw = (int)(rem % HWSZ);
  __builtin_prefetch(cls + e + 8192, 0, 1);     // global_prefetch_b8
  float x = cls[e];
  float c = ctr[n * HWSZ + hw];
  float s = sigf(x) * sigf(c);
  unsigned key = (x > PRE_LOGIT) ? __float_as_uint(s) : 0u;  // s>0 => monotone bits
  atomicAdd(&h[key >> 21], 1u);
  __syncthreads();
  for (int i = threadIdx.x; i < 2048; i += blockDim.x)
    if (h[i]) atomicAdd(&hist[n * 2048 + i], h[i]);
}

// ---------------- 2: pick per-image threshold bucket ------------------------
__global__ void k_thresh(const unsigned* __restrict__ hist,
                         unsigned* __restrict__ thresh) {
  __shared__ unsigned h[2048];
  int n = blockIdx.x;
  for (int i = threadIdx.x; i < 2048; i += blockDim.x) h[i] = hist[n * 2048 + i];
  __syncthreads();
  if (threadIdx.x == 0) {
    unsigned acc = 0; int T = 1;
    for (int t = 2047; t >= 1; --t) { acc += h[t]; if (acc >= KTOP) { T = t; break; } }
    thresh[n] = ((unsigned)T) << 21;   // lower bound of threshold bucket (>0)
  }
}

// ---------------- 3: compacting re-scan -------------------------------------
__global__ void k_collect(const float* __restrict__ cls,
                          const float* __restrict__ ctr,
                          const unsigned* __restrict__ thresh,
                          unsigned* __restrict__ cnt,
                          unsigned long long* __restrict__ list) {
  long long e = (long long)blockIdx.x * blockDim.x + threadIdx.x;
  const long long per = (long long)CC * HWSZ;
  int n  = (int)(e / per);
  long long rem = e - (long long)n * per;
  int c  = (int)(rem / HWSZ);
  int hw = (int)(rem % HWSZ);
  __builtin_prefetch(cls + e + 8192, 0, 1);
  float x  = cls[e];
  float ct = ctr[n * HWSZ + hw];
  float s  = sigf(x) * sigf(ct);
  unsigned key = (x > PRE_LOGIT) ? __float_as_uint(s) : 0u;
  if (key >= thresh[n]) {
    unsigned pos = atomicAdd(&cnt[n], 1u);
    if (pos < CAP) {
      unsigned idx = (unsigned)hw * CC + (unsigned)c;   // flat over [HW, C]
      list[(size_t)n * CAP + pos] =
          ((unsigned long long)key << 32) | (0xFFFFFFFFu - idx); // tie: low idx first
    }
  }
}

// ---------------- 4: per-image bitonic sort, keep top KP --------------------
__global__ __launch_bounds__(1024) void k_sort_topk(
    const unsigned long long* __restrict__ list,
    const unsigned* __restrict__ cnt,
    unsigned* __restrict__ topKey, unsigned* __restrict__ topIdx) {
  __shared__ unsigned long long s[CAP];  // 64 KB LDS (fine on CDNA5: 320 KB/WGP)
  int n = blockIdx.x;
  unsigned m = cnt[n]; if (m > CAP) m = CAP;
  for (int i = threadIdx.x; i < CAP; i += blockDim.x)
    s[i] = (i < (int)m) ? list[(size_t)n * CAP + i] : 0ull;
  __syncthreads();
  for (int kk = 2; kk <= CAP; kk <<= 1) {
    for (int j = kk >> 1; j > 0; j >>= 1) {
      for (int t = threadIdx.x; t < CAP; t += blockDim.x) {
        int ixj = t ^ j;
        if (ixj > t) {
          unsigned long long a = s[t], b = s[ixj];
          bool up = ((t & kk) == 0);
          if (up ? (a < b) : (a > b)) { s[t] = b; s[ixj] = a; }  // descending sort
        }
      }
      __syncthreads();
    }
  }
  for (int k = threadIdx.x; k < KP; k += blockDim.x) {
    unsigned long long e = s[k];
    topKey[n * KP + k] = (unsigned)(e >> 32);
    topIdx[n * KP + k] = 0xFFFFFFFFu - (unsigned)(e & 0xFFFFFFFFu);
  }
}

// ---------------- 5: decode boxes, offsets, areas ---------------------------
__global__ void k_decode(const unsigned* __restrict__ topKey,
                         const unsigned* __restrict__ topIdx,
                         const float* __restrict__ reg,
                         const float* __restrict__ loc,
                         const int* __restrict__ imsz,
                         float* __restrict__ boxes, float* __restrict__ boxOff,
                         float* __restrict__ area,  float* __restrict__ sc,
                         unsigned* __restrict__ labels, unsigned* __restrict__ validArr) {
  int g = blockIdx.x * blockDim.x + threadIdx.x;
  if (g >= NN * KP) return;
  int n = g / KP, k = g - n * KP;
  unsigned key = topKey[g];
  float hi = (float)imsz[2 * n + 0], wi = (float)imsz[2 * n + 1];
  if (k >= KTOP || key == 0u) {               // invalid / padding entry
    boxes[4*g+0] = 0.f; boxes[4*g+1] = 0.f; boxes[4*g+2] = 0.f; boxes[4*g+3] = 0.f;
    float fx = -1.0e8f + (float)g;            // isolated zero-area point
    boxOff[4*g+0] = fx;  boxOff[4*g+1] = 0.f; boxOff[4*g+2] = fx;  boxOff[4*g+3] = 0.f;
    area[g] = 0.f; sc[g] = 0.f; labels[g] = 0u; validArr[g] = 0u;
    return;
  }
  unsigned idx = topIdx[g];
  int li  = (int)(idx / CC);
  int lab = (int)(idx % CC) + 1;
  const float* rp = reg + (size_t)n * 4 * HWSZ;
  float l = rp[0 * HWSZ + li], t = rp[1 * HWSZ + li];
  float r = rp[2 * HWSZ + li], b = rp[3 * HWSZ + li];
  float x = loc[2 * li], y = loc[2 * li + 1];
  float x1 = fminf(fmaxf(x - l, 0.f), wi - 1.f);
  float x2 = fminf(fmaxf(x + r, 0.f), wi - 1.f);
  float y1 = fminf(fmaxf(y - t, 0.f), hi - 1.f);
  float y2 = fminf(fmaxf(y + b, 0.f), hi - 1.f);
  unsigned v = ((x2 - x1 >= 0.f) && (y2 - y1 >= 0.f)) ? 1u : 0u;
  boxes[4*g+0] = x1; boxes[4*g+1] = y1; boxes[4*g+2] = x2; boxes[4*g+3] = y2;
  float off = (float)lab * (fmaxf(wi, hi) + 1.f);   // multi-class NMS offset
  boxOff[4*g+0] = x1 + off; boxOff[4*g+1] = y1 + off;
  boxOff[4*g+2] = x2 + off; boxOff[4*g+3] = y2 + off;
  area[g]   = (x2 - x1) * (y2 - y1);
  sc[g]     = v ? sqrtf(__uint_as_float(key)) : 0.f;
  labels[g] = (unsigned)lab;
  validArr[g] = v;
}

// ---------------- 6: WMMA-tiled IoU suppression bitmask ---------------------
// One wave32 per 16x16 pair tile. Union base (area_i + area_j) for the whole
// tile comes from a single V_WMMA_F32_16X16X4_F32 rank-2 outer product:
//   A = [area_i | 1 | 0 | 0] (16x4), B = [1 ; area_j ; 0 ; 0] (4x16).
__global__ __launch_bounds__(256) void k_nms_mask(
    const float* __restrict__ boxOff, const float* __restrict__ area,
    unsigned* __restrict__ mask) {
  __shared__ float    sRow[8][16][4];
  __shared__ float    sCol[8][16][4];
  __shared__ float    sRA[8][16];
  __shared__ float    sCA[8][16];
  __shared__ unsigned sM[8][16];
  const int wv   = threadIdx.x >> 5;
  const int lane = threadIdx.x & 31;
  long long gw = (long long)blockIdx.x * 8 + wv;   // grid sized exactly: no guard
  const int TP = NTILE * NTILE;
  int n  = (int)(gw / TP);
  int tr = (int)(gw % TP);
  int ti = tr / NTILE, tj = tr - ti * NTILE;
  int i0 = ti * 16, j0 = tj * 16;
  const float* B = boxOff + (size_t)n * KP * 4;
  const float* A = area   + (size_t)n * KP;
  if (lane < 16) {
    sRow[wv][lane][0] = B[(i0 + lane) * 4 + 0];
    sRow[wv][lane][1] = B[(i0 + lane) * 4 + 1];
    sRow[wv][lane][2] = B[(i0 + lane) * 4 + 2];
    sRow[wv][lane][3] = B[(i0 + lane) * 4 + 3];
    sRA[wv][lane] = A[i0 + lane];
    sM[wv][lane]  = 0u;
  } else {
    int l2 = lane - 16;
    sCol[wv][l2][0] = B[(j0 + l2) * 4 + 0];
    sCol[wv][l2][1] = B[(j0 + l2) * 4 + 1];
    sCol[wv][l2][2] = B[(j0 + l2) * 4 + 2];
    sCol[wv][l2][3] = B[(j0 + l2) * 4 + 3];
    sCA[wv][l2] = A[j0 + l2];
  }
  __builtin_amdgcn_wave_barrier();   // wave-internal LDS is in-order; fence compiler
  // A-matrix 16x4 layout: lanes 0-15 M=0..15 {K0,K1}; lanes 16-31 {K2,K3}=0
  // B-matrix 4x16 layout: VGPR0 = K0 row (lanes 0-15) / K2 (16-31); VGPR1 = K1 / K3
  v2f a, b;
  a.x = (lane < 16) ? sRA[wv][lane & 15] : 0.f;   // A[M][0] = area_i
  a.y = (lane < 16) ? 1.f : 0.f;                  // A[M][1] = 1
  b.x = (lane < 16) ? 1.f : 0.f;                  // B[0][N] = 1
  b.y = (lane < 16) ? sCA[wv][lane & 15] : 0.f;   // B[1][N] = area_j
  v8f c = {};
  v8f d = __builtin_amdgcn_wmma_f32_16x16x4_f32(false, a, false, b,
                                                (short)0, c, false, false);
  // D layout: VGPR r -> M = r + 8*(lane>=16), N = lane & 15
  const int Nc    = lane & 15;
  const int mbase = (lane >> 4) << 3;
  float cx1 = sCol[wv][Nc][0], cy1 = sCol[wv][Nc][1];
  float cx2 = sCol[wv][Nc][2], cy2 = sCol[wv][Nc][3];
#pragma unroll
  for (int r2 = 0; r2 < 8; ++r2) {
    int M = mbase + r2;
    float rx1 = sRow[wv][M][0], ry1 = sRow[wv][M][1];
    float rx2 = sRow[wv][M][2], ry2 = sRow[wv][M][3];
    float ix = fmaxf(fminf(rx2, cx2) - fmaxf(rx1, cx1), 0.f);
    float iy = fmaxf(fminf(ry2, cy2) - fmaxf(ry1, cy1), 0.f);
    float inter = ix * iy;
    float uni = d[r2] - inter + 1e-9f;    // area_i + area_j - inter + eps (>0)
    if (inter > NMS_T * uni)              // iou > T without division
      atomicOr(&sM[wv][M], 1u << Nc);
  }
  __builtin_amdgcn_wave_barrier();
  if (lane < 16) {
    unsigned m = sM[wv][lane];
    if (m) {
      int row = i0 + lane;
      atomicOr(&mask[((size_t)n * KP + row) * 32 + (j0 >> 5)], m << (j0 & 31));
    }
  }
}

// ---------------- 7: sequential greedy suppress (one wave per image) --------
__global__ void k_suppress(const unsigned* __restrict__ mask,
                           const unsigned* __restrict__ validArr,
                           const float* __restrict__ sc,
                           float* __restrict__ fscore) {
  __shared__ unsigned sup[32];
  int n = blockIdx.x, t = threadIdx.x;   // blockDim = 32
  unsigned vb = 0u;
  for (int b2 = 0; b2 < 32; ++b2) {
    int i = t * 32 + b2;
    if (i < KP && validArr[n * KP + i]) vb |= 1u << b2;
  }
  sup[t] = ~vb;                           // sup init = ~valid
  __syncthreads();
  for (int i = 0; i < KTOP; ++i) {
    unsigned sw = sup[i >> 5];
    if (!((sw >> (i & 31)) & 1u)) {       // row i not suppressed -> apply it
      unsigned m = mask[((size_t)n * KP + i) * 32 + t];
      int iw = i >> 5, bp = i & 31;
      if (t < iw) m = 0u;
      else if (t == iw) m &= (bp == 31) ? 0u : ~((1u << (bp + 1)) - 1u); // only j > i
      atomicOr(&sup[t], m);
    }
    __syncthreads();
  }
  for (int k = t; k < KP; k += 32) {
    bool su = (sup[k >> 5] >> (k & 31)) & 1u;
    bool v  = validArr[n * KP + k] != 0u;
    fscore[n * KP + k] = (!su && v) ? sc[n * KP + k] : -1.0f;
  }
}

// ---------------- 8: final per-image top-100 + emit -------------------------
__global__ __launch_bounds__(512) void k_final(
    const float* __restrict__ fscore, const float* __restrict__ boxes,
    const unsigned* __restrict__ labels, float* __restrict__ out) {
  __shared__ unsigned long long s[1024];
  int n = blockIdx.x;
  for (int i = threadIdx.x; i < 1024; i += blockDim.x) {
    if (i < KP) {
      float f = fscore[n * KP + i];                 // in {-1} U [0,1]
      unsigned key = __float_as_uint(f + 2.0f);     // positive -> monotone bits
      s[i] = ((unsigned long long)key << 32) | (0xFFFFFFFFu - (unsigned)i);
    } else s[i] = 0ull;
  }
  __syncthreads();
  for (int kk = 2; kk <= 1024; kk <<= 1) {
    for (int j = kk >> 1; j > 0; j >>= 1) {
      for (int t = threadIdx.x; t < 1024; t += blockDim.x) {
        int ixj = t ^ j;
        if (ixj > t) {
          unsigned long long a = s[t], b = s[ixj];
          bool up = ((t & kk) == 0);
          if (up ? (a < b) : (a > b)) { s[t] = b; s[ixj] = a; }
        }
      }
      __syncthreads();
    }
  }
  for (int p = threadIdx.x; p < 100; p += blockDim.x) {
    unsigned long long e = s[p];
    unsigned k = 0xFFFFFFFFu - (unsigned)(e & 0xFFFFFFFFu);
    float fs = (k < KP) ? fscore[n * KP + k] : -1.0f;
    bool fv = fs >= 0.f;
    size_t ob = ((size_t)n * 100 + p) * 5;
    const float* bx = boxes + ((size_t)n * KP + (k < KP ? k : 0)) * 4;
    out[ob + 0] = fv ? bx[0] : 0.f;
    out[ob + 1] = fv ? bx[1] : 0.f;
    out[ob + 2] = fv ? bx[2] : 0.f;
    out[ob + 3] = fv ? bx[3] : 0.f;
    out[ob + 4] = fv ? fs    : 0.f;
    out[OUT_DETS + n * 100 + p] = fv ? (float)labels[n * KP + (k < KP ? k : 0)] : 0.f;
  }
}

// ---------------- host-side launch ------------------------------------------
extern "C" void kernel_launch(void* const* d_in, const int* in_sizes, int n_in,
                              void* d_out, int out_size, void* d_ws, size_t ws_size,
                              hipStream_t stream) {
  const float* loc  = (const float*)d_in[0];   // [HW,2]
  const float* cls  = (const float*)d_in[1];   // [N,C,H,W]
  const float* reg  = (const float*)d_in[2];   // [N,4,H,W]
  const float* ctr  = (const float*)d_in[3];   // [N,1,H,W]
  const int*   ims  = (const int*)  d_in[4];   // [N,2]
  float* out = (float*)d_out;
  char* ws = (char*)d_ws;

  size_t o = 0;
  auto alloc = [&](size_t bytes) { size_t r = o; o += bytes; o = (o + 255) & ~(size_t)255; return r; };
  // zeroed region first (hist + cnt + mask contiguous):
  size_t o_hist = alloc((size_t)NN * 2048 * 4);
  size_t o_cnt  = alloc((size_t)NN * 4);
  size_t o_mask = alloc((size_t)NN * KP * 32 * 4);
  size_t zeroBytes = o;                       // zero everything up to here
  size_t o_thr  = alloc((size_t)NN * 4);
  size_t o_list = alloc((size_t)NN * CAP * 8);
  size_t o_tkey = alloc((size_t)NN * KP * 4);
  size_t o_tidx = alloc((size_t)NN * KP * 4);
  size_t o_box  = alloc((size_t)NN * KP * 4 * 4);
  size_t o_boff = alloc((size_t)NN * KP * 4 * 4);
  size_t o_area = alloc((size_t)NN * KP * 4);
  size_t o_sc   = alloc((size_t)NN * KP * 4);
  size_t o_lab  = alloc((size_t)NN * KP * 4);
  size_t o_val  = alloc((size_t)NN * KP * 4);
  size_t o_fsc  = alloc((size_t)NN * KP * 4);
  (void)ws_size; (void)n_in; (void)in_sizes; (void)out_size;

  unsigned* hist = (unsigned*)(ws + o_hist);
  unsigned* cnt  = (unsigned*)(ws + o_cnt);
  unsigned* msk  = (unsigned*)(ws + o_mask);
  unsigned* thr  = (unsigned*)(ws + o_thr);
  unsigned long long* lst = (unsigned long long*)(ws + o_list);
  unsigned* tkey = (unsigned*)(ws + o_tkey);
  unsigned* tidx = (unsigned*)(ws + o_tidx);
  float* box  = (float*)(ws + o_box);
  float* boff = (float*)(ws + o_boff);
  float* ar   = (float*)(ws + o_area);
  float* sc   = (float*)(ws + o_sc);
  unsigned* lab = (unsigned*)(ws + o_lab);
  unsigned* val = (unsigned*)(ws + o_val);
  float* fsc  = (float*)(ws + o_fsc);

  long long zc = (long long)(zeroBytes / 4);
  k_zero<<<(unsigned)((zc + 255) / 256), 256, 0, stream>>>((unsigned*)ws, zc);

  const int scanBlocks = (int)(TOTAL / 256);              // 76000, exact
  k_scan_hist<<<scanBlocks, 256, 0, stream>>>(cls, ctr, hist);
  k_thresh<<<NN, 256, 0, stream>>>(hist, thr);
  k_collect<<<scanBlocks, 256, 0, stream>>>(cls, ctr, thr, cnt, lst);
  k_sort_topk<<<NN, 1024, 0, stream>>>(lst, cnt, tkey, tidx);
  k_decode<<<(NN * KP + 255) / 256, 256, 0, stream>>>(tkey, tidx, reg, loc, ims,
                                                      box, boff, ar, sc, lab, val);
  const int tiles = NN * NTILE * NTILE;                   // 63504 = 7938 * 8
  k_nms_mask<<<tiles / 8, 256, 0, stream>>>(boff, ar, msk);
  k_suppress<<<NN, 32, 0, stream>>>(msk, val, sc, fsc);
  k_final<<<NN, 512, 0, stream>>>(fsc, box, lab, out);
}